// ClassCapsule_996432412895
// MI455X (gfx1250) — compile-verified
//
#include <hip/hip_runtime.h>

typedef __attribute__((ext_vector_type(2))) float v2f;
typedef __attribute__((ext_vector_type(8))) float v8f;

#define Bn 512
#define Rn 1152
#define Cn 10
#define On 16
#define In 8
#define BT 16          // batch tile = WMMA N
#define NTHREADS 512
#define NWAVES 16
#define RITERS 3

__global__ __launch_bounds__(NTHREADS)
void caps_routing_kernel(const float* __restrict__ x,
                         const float* __restrict__ W,
                         float* __restrict__ out) {
  // s_j accumulator and running sum of v (for implicit b_ij = u_hat . Vsum)
  __shared__ float sJ[Cn][On][BT];
  __shared__ float sV[Cn][On][BT];

  const int tid  = threadIdx.x;
  const int lane = tid & 31;
  const int wave = tid >> 5;
  const int bl   = lane & 15;   // batch within tile (WMMA N index)
  const int hi   = lane >> 4;   // 1 for lanes 16..31 (K/M upper halves)
  const int b0   = blockIdx.x * BT;

  for (int i = tid; i < Cn * On * BT; i += NTHREADS) {
    ((float*)sJ)[i] = 0.f;
    ((float*)sV)[i] = 0.f;
  }
  __syncthreads();

  for (int it = 0; it < RITERS; ++it) {
    v8f acc[Cn] = {};           // WMMA C/D: s_j partial, per class tile
    float vs[Cn][8];            // per-lane slice of Vsum: o = j + hi*8, b = bl
#pragma unroll
    for (int c = 0; c < Cn; ++c)
#pragma unroll
      for (int j = 0; j < 8; ++j)
        vs[c][j] = sV[c][j + hi * 8][bl];

    for (int r = wave; r < Rn; r += NWAVES) {
      // B operand (x): VGPR0 = K even, VGPR1 = K odd; lanes>=16 hold K+2
      const float* xp = x + ((size_t)(b0 + bl) * Rn + r) * In + hi * 2;
      v2f bx0 = *(const v2f*)(xp);       // i = hi*2, hi*2+1      (K-half 0)
      v2f bx1 = *(const v2f*)(xp + 4);   // i = 4+hi*2, 4+hi*2+1  (K-half 1)

      const float* wr = W + (size_t)r * Cn * On * In;
      __builtin_prefetch(wr + (size_t)NWAVES * Cn * On * In, 0, 0);

      float cscale[Cn];
      if (it == 0) {
        // softmax of zero logits == exactly 1/C
#pragma unroll
        for (int c = 0; c < Cn; ++c) cscale[c] = 0.1f;
      } else {
        // Pass 1: u_hat tiles -> agreement logits = u_hat . Vsum
        float logit[Cn];
#pragma unroll
        for (int c = 0; c < Cn; ++c) {
          const float* wp = wr + ((size_t)c * On + bl) * In + hi * 2;
          v2f a0 = *(const v2f*)(wp);
          v2f a1 = *(const v2f*)(wp + 4);
          v8f z = {};
          v8f u = __builtin_amdgcn_wmma_f32_16x16x4_f32(
              false, a0, false, bx0, (short)0, z, false, false);
          u = __builtin_amdgcn_wmma_f32_16x16x4_f32(
              false, a1, false, bx1, (short)0, u, false, false);
          float d = 0.f;
#pragma unroll
          for (int j = 0; j < 8; ++j) d += u[j] * vs[c][j];
          d += __shfl_xor(d, 16, 32);   // combine o halves (lanes l <-> l+16)
          logit[c] = d;
        }
        // per-lane softmax over classes
        float m = logit[0];
#pragma unroll
        for (int c = 1; c < Cn; ++c) m = fmaxf(m, logit[c]);
        float ssum = 0.f;
#pragma unroll
        for (int c = 0; c < Cn; ++c) {
          cscale[c] = __expf(logit[c] - m);
          ssum += cscale[c];
        }
        float inv = 1.0f / ssum;
#pragma unroll
        for (int c = 0; c < Cn; ++c) cscale[c] *= inv;
      }

      // Pass 2: s_j[c] += A_c x (B * c_ij)  — accumulates in WMMA C across r
#pragma unroll
      for (int c = 0; c < Cn; ++c) {
        const float* wp = wr + ((size_t)c * On + bl) * In + hi * 2;
        v2f a0 = *(const v2f*)(wp);
        v2f a1 = *(const v2f*)(wp + 4);
        v2f bc0 = bx0 * cscale[c];
        v2f bc1 = bx1 * cscale[c];
        acc[c] = __builtin_amdgcn_wmma_f32_16x16x4_f32(
            false, a0, false, bc0, (short)0, acc[c], false, false);
        acc[c] = __builtin_amdgcn_wmma_f32_16x16x4_f32(
            false, a1, false, bc1, (short)0, acc[c], false, false);
      }
    }

    // Deterministic cross-wave reduction (fixed wave order, no FP atomics)
    for (int w = 0; w < NWAVES; ++w) {
      if (wave == w) {
#pragma unroll
        for (int c = 0; c < Cn; ++c)
#pragma unroll
          for (int j = 0; j < 8; ++j)
            sJ[c][j + hi * 8][bl] += acc[c][j];
      }
      __syncthreads();
    }

    // Squash: one thread per (class, batch)
    if (tid < Cn * BT) {
      const int c = tid / BT;
      const int b = tid % BT;
      float s[On];
      float sq = 0.f;
#pragma unroll
      for (int o = 0; o < On; ++o) {
        s[o] = sJ[c][o][b];
        sq += s[o] * s[o];
      }
      const float sc = sq / ((1.f + sq) * sqrtf(sq + 1e-8f));
      if (it < RITERS - 1) {
#pragma unroll
        for (int o = 0; o < On; ++o) sV[c][o][b] += sc * s[o];
      } else {
#pragma unroll
        for (int o = 0; o < On; ++o)
          out[((size_t)(b0 + b) * Cn + c) * On + o] = sc * s[o];
      }
    }
    __syncthreads();
    if (it < RITERS - 1) {
      for (int i = tid; i < Cn * On * BT; i += NTHREADS) ((float*)sJ)[i] = 0.f;
      __syncthreads();
    }
  }
}

extern "C" void kernel_launch(void* const* d_in, const int* in_sizes, int n_in,
                              void* d_out, int out_size, void* d_ws, size_t ws_size,
                              hipStream_t stream) {
  const float* x = (const float*)d_in[0];   // (B, R, I) f32
  const float* W = (const float*)d_in[1];   // (R, C, O, I) f32
  float* out = (float*)d_out;               // (B, C, O) f32
  (void)in_sizes; (void)n_in; (void)out_size; (void)d_ws; (void)ws_size;
  dim3 grid(Bn / BT);
  dim3 block(NTHREADS);
  hipLaunchKernelGGL(caps_routing_kernel, grid, block, 0, stream, x, W, out);
}